// Foo_49100066128435
// MI455X (gfx1250) — compile-verified
//
#include <hip/hip_runtime.h>
#include <cstdint>

// ---------------------------------------------------------------------------
// HashMap.get on MI455X (gfx1250).
// Keys are a permutation of the even values in [0, 2N): a perfect
// direct-address table inv[k>>1] = i answers every query with one gather.
// The 40MB int32 table lives in d_ws and stays resident in the 192MB L2:
// every streaming access (keys, queries, outputs, async staging) is tagged
// non-temporal so only the table occupies L2. Query tiles are staged
// global->LDS with gfx1250 async copies (ASYNCcnt), double buffered, so the
// stream load latency hides behind the random table gathers.
// Roofline: ~440MB of HBM traffic @ 23.3 TB/s => ~19us floor; gathers hit L2.
// ---------------------------------------------------------------------------

typedef long long ll2 __attribute__((ext_vector_type(2)));

#define TPB    256
#define QPT    8                    // queries per thread per tile (gather MLP)
#define TILE_Q (TPB * QPT)          // 2048 queries per block per stage
#define LOOKUP_BLOCKS 2048

// Async-copy this thread's QPT int64 queries (64B) into LDS: four b128 async
// ops. INST_OFFSET is added to BOTH the global and LDS address by the ISA,
// so one base address pair serves all four copies. NT hint keeps the query
// stream out of L2 (table residency).
__device__ __forceinline__ void async_stage_queries(const long long* gsrc,
                                                    const long long* lds_dst) {
    unsigned            lds = (unsigned)(uintptr_t)lds_dst;        // low 32b of flat
    unsigned long long  g   = (unsigned long long)(uintptr_t)gsrc; // = LDS offset
    asm volatile("global_load_async_to_lds_b128 %0, %1, off th:TH_LOAD_NT"
                 :: "v"(lds), "v"(g) : "memory");
    asm volatile("global_load_async_to_lds_b128 %0, %1, off offset:16 th:TH_LOAD_NT"
                 :: "v"(lds), "v"(g) : "memory");
    asm volatile("global_load_async_to_lds_b128 %0, %1, off offset:32 th:TH_LOAD_NT"
                 :: "v"(lds), "v"(g) : "memory");
    asm volatile("global_load_async_to_lds_b128 %0, %1, off offset:48 th:TH_LOAD_NT"
                 :: "v"(lds), "v"(g) : "memory");
}

// ---- Kernel 1: build perfect table. inv[key[i]>>1] = i --------------------
// Keys streamed with NT b128 loads (read once, don't pollute L2); scatter
// stores use default RT policy so the table is left hot in L2.
__global__ void __launch_bounds__(TPB)
build_inv_kernel(const long long* __restrict__ key, int* __restrict__ inv, int n) {
    const int i = (blockIdx.x * blockDim.x + threadIdx.x) * 2;
    if (i + 1 < n) {
        ll2 k = __builtin_nontemporal_load((const ll2*)(key + i));
        inv[(int)(k.x >> 1)] = i;
        inv[(int)(k.y >> 1)] = i + 1;
    } else if (i < n) {
        long long k = __builtin_nontemporal_load(key + i);
        inv[(int)(k >> 1)] = i;
    }
}

// ---- Kernel 2: lookup -----------------------------------------------------
__global__ void __launch_bounds__(TPB)
lookup_kernel(const long long* __restrict__ query,
              const int* __restrict__ inv,
              long long* __restrict__ out,
              int n_q, int n_keys) {
    __shared__ long long qbuf[2][TILE_Q];
    const int tid = threadIdx.x;
    const unsigned long long two_n = 2ull * (unsigned long long)n_keys;
    const int fullTiles = n_q / TILE_Q;

    int tile  = blockIdx.x;
    int stage = 0;
    if (tile < fullTiles) {
        async_stage_queries(query + (size_t)tile * TILE_Q + tid * QPT,
                            &qbuf[0][tid * QPT]);
    }

    while (tile < fullTiles) {
        const int next = tile + gridDim.x;
        if (next < fullTiles) {
            // Prefetch next tile into the other buffer, then release the
            // current stage: async loads complete in order, so asynccnt<=4
            // means the 4 older (current-stage) copies have landed in LDS.
            async_stage_queries(query + (size_t)next * TILE_Q + tid * QPT,
                                &qbuf[stage ^ 1][tid * QPT]);
            asm volatile("s_wait_asynccnt 0x4" ::: "memory");
        } else {
            asm volatile("s_wait_asynccnt 0x0" ::: "memory");
        }

        // Pull this thread's 8 queries from LDS as four b128 reads.
        long long q[QPT];
#pragma unroll
        for (int j = 0; j < QPT; j += 2) {
            ll2 v = *(const ll2*)&qbuf[stage][tid * QPT + j];
            q[j] = v.x; q[j + 1] = v.y;
        }

        // Branch-free: miss lanes gather slot 0 (hot in L2) and get masked.
        // 8 independent gathers per thread stay in flight together.
        long long r[QPT];
#pragma unroll
        for (int j = 0; j < QPT; ++j) {
            const bool hit = ((unsigned long long)q[j] < two_n) && ((q[j] & 1) == 0);
            const long long slot = hit ? (q[j] >> 1) : 0;
            const int idx = inv[slot];          // RT load: L2-resident table
            r[j] = hit ? (long long)idx : -1ll;
        }

        long long* obase = out + (size_t)tile * TILE_Q + (size_t)tid * QPT;
#pragma unroll
        for (int j = 0; j < QPT; j += 2) {
            ll2 v; v.x = r[j]; v.y = r[j + 1];
            __builtin_nontemporal_store(v, (ll2*)(obase + j));
        }

        stage ^= 1;
        tile = next;
    }

    // Tail (n_q not divisible by TILE_Q): plain NT-load path.
    for (int i = fullTiles * TILE_Q + blockIdx.x * blockDim.x + tid;
         i < n_q; i += gridDim.x * blockDim.x) {
        const long long qq = __builtin_nontemporal_load(&query[i]);
        const bool hit = ((unsigned long long)qq < two_n) && ((qq & 1) == 0);
        const long long slot = hit ? (qq >> 1) : 0;
        const int idx = inv[slot];
        __builtin_nontemporal_store(hit ? (long long)idx : -1ll, &out[i]);
    }
}

extern "C" void kernel_launch(void* const* d_in, const int* in_sizes, int n_in,
                              void* d_out, int out_size, void* d_ws, size_t ws_size,
                              hipStream_t stream) {
    const long long* key   = (const long long*)d_in[0];
    const long long* query = (const long long*)d_in[1];
    long long*       out   = (long long*)d_out;
    int*             inv   = (int*)d_ws;       // N_KEYS * 4B = 40MB scratch

    const int n_keys = in_sizes[0];
    const int n_q    = in_sizes[1];

    const int bblocks = (n_keys + 2 * TPB - 1) / (2 * TPB);
    build_inv_kernel<<<bblocks, TPB, 0, stream>>>(key, inv, n_keys);
    lookup_kernel<<<LOOKUP_BLOCKS, TPB, 0, stream>>>(query, inv, out, n_q, n_keys);
}